// ConvAttention_70841190580679
// MI455X (gfx1250) — compile-verified
//
#include <hip/hip_runtime.h>
#include <math.h>

typedef __attribute__((ext_vector_type(2))) float v2f;
typedef __attribute__((ext_vector_type(4))) float f4;
typedef __attribute__((ext_vector_type(8))) float v8f;

#define CSTRIDE 65536     // channel stride in x/out  (h*w*s = 64*64*16)
#define BSTRIDE 4194304   // batch stride             (c*h*w*s)
#define SB_OFF  3200      // ws: folded conv biases (2 floats)
#define SQ_OFF  4096      // ws: sq map  (4*64*64*16 floats)
#define SK_OFF  (4096 + 262144)

// ---------------------------------------------------------------------------
// Kernel 1: effective conv weights.  wq_eff[tap][c'] = sum_c w2q[c,tap]*w1q[c,c']
// (conv(linear(x)) == conv_eff(x); bias folds to a spatial constant).
// ---------------------------------------------------------------------------
__global__ void ca_eff_weights(const float* __restrict__ w1,
                               const float* __restrict__ b1,
                               const float* __restrict__ w2,
                               float* __restrict__ ws) {
    int t = blockIdx.x * 256 + threadIdx.x;
    if (t < 3200) {
        int qk = t / 1600, rem = t % 1600;
        int tap = rem / 64, cp = rem % 64;
        float acc = 0.f;
        #pragma unroll 8
        for (int c = 0; c < 64; ++c)
            acc += w2[(qk * 64 + c) * 25 + tap] * w1[(qk * 64 + c) * 64 + cp];
        ws[t] = acc;                       // [0..1599]=wq_eff, [1600..3199]=wk_eff
    } else if (t == 3200 || t == 3201) {
        int off = (t - 3200) * 64;
        float acc = 0.f;
        for (int c = 0; c < 64; ++c) {
            float bv = b1[off + c];
            for (int tap = 0; tap < 25; ++tap)
                acc += w2[(off + c) * 25 + tap] * bv;
        }
        ws[SB_OFF + (t - 3200)] = acc;     // folded sq / sk bias
    }
}

// ---------------------------------------------------------------------------
// Kernel 2: sq/sk = 5x5 reflect conv of x with the effective weights.
// One thread = one (pixel, 4-s chunk); weights staged in LDS; float4 loads.
// ---------------------------------------------------------------------------
__global__ void ca_conv_scores(const float* __restrict__ x,
                               const float* __restrict__ ws,
                               float* __restrict__ sq,
                               float* __restrict__ sk) {
    __shared__ float lwq[1600];
    __shared__ float lwk[1600];
    for (int i = threadIdx.x; i < 1600; i += 256) {
        lwq[i] = ws[i];
        lwk[i] = ws[1600 + i];
    }
    __syncthreads();

    int gid = blockIdx.x * 256 + threadIdx.x;
    int p  = gid >> 2;
    int s4 = (gid & 3) * 4;
    int b = p >> 12, hw = p & 4095, h = hw >> 6, w = hw & 63;
    const float* xb = x + (long)b * BSTRIDE + s4;

    f4 aq = {0.f, 0.f, 0.f, 0.f};
    f4 ak = {0.f, 0.f, 0.f, 0.f};
    for (int dy = -2; dy <= 2; ++dy) {
        int hy = h + dy; hy = hy < 0 ? -hy : hy; hy = hy > 63 ? 126 - hy : hy;
        for (int dx = -2; dx <= 2; ++dx) {
            int wx = w + dx; wx = wx < 0 ? -wx : wx; wx = wx > 63 ? 126 - wx : wx;
            int tap = (dy + 2) * 5 + (dx + 2);
            const float* xp = xb + hy * 1024 + wx * 16;
            #pragma unroll 8
            for (int c = 0; c < 64; ++c) {
                f4 xv = *(const f4*)(xp + c * CSTRIDE);
                float wqv = lwq[tap * 64 + c];
                float wkv = lwk[tap * 64 + c];
                aq += wqv * xv;
                ak += wkv * xv;
            }
        }
    }
    aq += ws[SB_OFF + 0];
    ak += ws[SB_OFF + 1];
    *(f4*)(sq + p * 16 + s4) = aq;
    *(f4*)(sk + p * 16 + s4) = ak;
}

// ---------------------------------------------------------------------------
// Kernel 3: V projection via f32 WMMA.  Per pixel: V(64x16) = w1v(64x64) @ x(64x16).
// A = w1v (LDS, b64 fragments), B = x (direct global loads in WMMA layout),
// D rows are contiguous s-rows -> coalesced 64B stores into d_out (V parked there).
// ---------------------------------------------------------------------------
__global__ void ca_vproj(const float* __restrict__ x,
                         const float* __restrict__ w1,
                         const float* __restrict__ b1,
                         float* __restrict__ out) {
    __shared__ float lw[4096];   // w1v: [c_out][c_in]
    __shared__ float lb[64];     // b1v
    {
        f4* lw4 = (f4*)lw;
        const f4* w1v4 = (const f4*)(w1 + 128 * 64);
        for (int i = threadIdx.x; i < 1024; i += 256) lw4[i] = w1v4[i];
        if (threadIdx.x < 64) lb[threadIdx.x] = b1[128 + threadIdx.x];
    }
    __syncthreads();

    int wave = threadIdx.x >> 5;
    int lane = threadIdx.x & 31;
    int hi = lane >> 4;          // lane half selects K/M sub-tiles
    int ls = lane & 15;
    int p = blockIdx.x * 8 + wave;
    long base = (long)(p >> 12) * BSTRIDE + (long)(p & 4095) * 16;
    const float* xb = x + base;
    float* ob = out + base;

    // B fragments: B(4x16) step k: VGPR0 = rows {4k | 4k+2}, VGPR1 = {4k+1 | 4k+3}
    v2f bx[16];
    #pragma unroll
    for (int k = 0; k < 16; ++k) {
        int cB = 4 * k + 2 * hi;
        v2f bb;
        bb.x = xb[cB * CSTRIDE + ls];
        bb.y = xb[(cB + 1) * CSTRIDE + ls];
        bx[k] = bb;
    }

    #pragma unroll
    for (int n = 0; n < 4; ++n) {          // 4 blocks of 16 output channels
        v8f acc;
        #pragma unroll
        for (int r = 0; r < 8; ++r)        // C preloaded with bias (per-c constant)
            acc[r] = lb[n * 16 + r + 8 * hi];
        #pragma unroll
        for (int k = 0; k < 16; ++k) {     // K = 64 in steps of 4
            int row = n * 16 + ls;
            int col = 4 * k + 2 * hi;
            v2f a = *(const v2f*)(&lw[row * 64 + col]);   // A: contiguous K pair
            acc = __builtin_amdgcn_wmma_f32_16x16x4_f32(
                false, a, false, bx[k], (short)0, acc, false, false);
        }
        #pragma unroll
        for (int r = 0; r < 8; ++r)        // each half-wave writes one 64B c-row
            ob[(n * 16 + r + 8 * hi) * CSTRIDE + ls] = acc[r];
    }
}

// ---------------------------------------------------------------------------
// Kernel 4: per-pixel softmax + out = V @ attn^T via f32 WMMA, in place over d_out.
// Every lane owns attn row s=lane&15 in registers -> B fragments are reg selects.
// ---------------------------------------------------------------------------
__global__ void ca_attn(const float* __restrict__ ws,
                        const float* __restrict__ b2,
                        float* __restrict__ out) {
    int wave = threadIdx.x >> 5;
    int lane = threadIdx.x & 31;
    int hi = lane >> 4;
    int ls = lane & 15;
    int p = blockIdx.x * 8 + wave;
    long base = (long)(p >> 12) * BSTRIDE + (long)(p & 4095) * 16;
    float* ob = out + base;

    const float* sqp = ws + SQ_OFF + p * 16;
    const float* skp = ws + SK_OFF + p * 16;
    float b2v = b2[0];
    float sqv = sqp[ls];

    float row[16];
    float m = -1e30f;
    #pragma unroll
    for (int t = 0; t < 16; ++t) {
        row[t] = sqv + skp[t] + b2v;
        m = fmaxf(m, row[t]);
    }
    float sum = 0.f;
    #pragma unroll
    for (int t = 0; t < 16; ++t) { row[t] = __expf(row[t] - m); sum += row[t]; }
    float inv = 1.0f / sum;
    #pragma unroll
    for (int t = 0; t < 16; ++t) row[t] *= inv;   // attn[s=ls][t]

    // Preload ALL V fragments before any store (in-place safety).
    v2f af[4][4];
    #pragma unroll
    for (int n = 0; n < 4; ++n)
        #pragma unroll
        for (int k = 0; k < 4; ++k)
            af[n][k] = *(const v2f*)(ob + (n * 16 + ls) * CSTRIDE + 4 * k + 2 * hi);

    #pragma unroll
    for (int n = 0; n < 4; ++n) {
        v8f acc = {};
        #pragma unroll
        for (int k = 0; k < 4; ++k) {      // K = t = 16 in steps of 4
            v2f bb;                         // B[t][s] = attn[s][t]
            bb.x = hi ? row[4 * k + 2] : row[4 * k];
            bb.y = hi ? row[4 * k + 3] : row[4 * k + 1];
            acc = __builtin_amdgcn_wmma_f32_16x16x4_f32(
                false, af[n][k], false, bb, (short)0, acc, false, false);
        }
        #pragma unroll
        for (int r = 0; r < 8; ++r)
            ob[(n * 16 + r + 8 * hi) * CSTRIDE + ls] = acc[r];
    }
}

// ---------------------------------------------------------------------------
extern "C" void kernel_launch(void* const* d_in, const int* in_sizes, int n_in,
                              void* d_out, int out_size, void* d_ws, size_t ws_size,
                              hipStream_t stream) {
    const float* x  = (const float*)d_in[0];
    const float* w1 = (const float*)d_in[1];
    const float* b1 = (const float*)d_in[2];
    const float* w2 = (const float*)d_in[3];
    const float* b2 = (const float*)d_in[4];
    float* out = (float*)d_out;
    float* wsf = (float*)d_ws;

    ca_eff_weights<<<13, 256, 0, stream>>>(w1, b1, w2, wsf);
    ca_conv_scores<<<256, 256, 0, stream>>>(x, wsf, wsf + SQ_OFF, wsf + SK_OFF);
    ca_vproj<<<2048, 256, 0, stream>>>(x, w1, b1, out);
    ca_attn<<<2048, 256, 0, stream>>>(wsf, b2, out);
}